// StaticEncoder_45037027066546
// MI455X (gfx1250) — compile-verified
//
#include <hip/hip_runtime.h>

// ---------------------------------------------------------------------------
// StaticEncoder for MI455X (gfx1250): fused skeleton-MLP via V_WMMA_F32_16X16X4_F32
//
//   out1 = leaky( x @ M1^T + c1 ),  M1 = pool_w @ (W1*mask1)   [84x84]
//   out2 = leaky( out1 @ A2^T + b2), A2 = W2*mask2             [168x84]
//   d_out = [ x (B*84) | out1 (B*84) | out2 (B*168) ]
//
// Memory-bound problem (~440MB moved, ~11 GFLOP) -> fp32 WMMA keeps exact
// reference precision at zero cost. Weights are prepped once into d_ws in
// WMMA-B-fragment-swizzled order so the hot loop is ds_load_b64 + v_wmma.
// ---------------------------------------------------------------------------

typedef float v2f __attribute__((ext_vector_type(2)));
typedef float v8f __attribute__((ext_vector_type(8)));

#define IN_F   84     // J*C
#define H1_F   168    // J*2C (pre-pool width, only used in prep reduction)
#define P_F    84     // J2*2C = out1 width
#define OUT_F  168    // J2*4C = out2 width
#define KSTEPS 21     // 84 / 4  (K dim of both GEMMs, exact)
#define NT1    6      // N tiles GEMM1: 84 -> 96
#define NT2    11     // N tiles GEMM2: 168 -> 176
#define N1PAD  96
#define N2PAD  176

// swizzled weight layout: [tile][kstep][lane] * float2  (lane L<16: K=4k..4k+1,
// N=16t+L ; lane L+16: K=4k+2..4k+3, N=16t+L) -> one ds_load_b64 per WMMA step.
#define W1S_FLOATS (NT1 * KSTEPS * 64)            // 8064
#define W2S_FLOATS (NT2 * KSTEPS * 64)            // 14784
#define C1_OFF     (W1S_FLOATS + W2S_FLOATS)      // 22848
#define B2_OFF     (C1_OFF + N1PAD)               // 22944
#define WS_FLOATS  (B2_OFF + N2PAD)               // 23120 floats = 92480 B

#define STAGE_STRIDE 98                            // even (b64 align) + conflict-free
#define STAGE_FLOATS (16 * STAGE_STRIDE)           // 1568
#define WAVES 8
#define CHARS_PER_BLOCK (WAVES * 16)
#define SMEM_FLOATS (WS_FLOATS + WAVES * STAGE_FLOATS)   // 35664 -> 142656 B

__device__ __forceinline__ float leaky(float v) {
    return v >= 0.0f ? v : 0.2f * v;
}

// ---------------------------------------------------------------------------
// Prep: fuse pool into layer1, apply masks, swizzle into B-fragment order.
// Tiny (<2 MFLOP); grid-stride over the 23120 workspace floats.
// ---------------------------------------------------------------------------
__global__ void enc_prep_kernel(const float* __restrict__ W1, const float* __restrict__ b1,
                                const float* __restrict__ W2, const float* __restrict__ b2,
                                const float* __restrict__ mask1, const float* __restrict__ mask2,
                                const float* __restrict__ pool_w, float* __restrict__ ws) {
    for (int idx = blockIdx.x * blockDim.x + threadIdx.x; idx < WS_FLOATS;
         idx += gridDim.x * blockDim.x) {
        float val = 0.0f;
        if (idx < W1S_FLOATS) {
            int pair  = idx >> 1, comp = idx & 1;
            int lane  = pair & 31;
            int kstep = (pair >> 5) % KSTEPS;
            int t     = pair / (32 * KSTEPS);
            int N     = 16 * t + (lane & 15);
            int K     = 4 * kstep + ((lane >> 4) << 1) + comp;
            if (N < P_F) {
                float s = 0.0f;
                for (int j = 0; j < H1_F; ++j)
                    s += pool_w[N * H1_F + j] * W1[j * IN_F + K] * mask1[j * IN_F + K];
                val = s;
            }
        } else if (idx < C1_OFF) {
            int i2    = idx - W1S_FLOATS;
            int pair  = i2 >> 1, comp = i2 & 1;
            int lane  = pair & 31;
            int kstep = (pair >> 5) % KSTEPS;
            int t     = pair / (32 * KSTEPS);
            int N     = 16 * t + (lane & 15);
            int K     = 4 * kstep + ((lane >> 4) << 1) + comp;
            if (N < OUT_F) val = W2[N * P_F + K] * mask2[N * P_F + K];
        } else if (idx < B2_OFF) {
            int n = idx - C1_OFF;
            if (n < P_F) {
                float s = 0.0f;
                for (int j = 0; j < H1_F; ++j) s += pool_w[n * H1_F + j] * b1[j];
                val = s;
            }
        } else {
            int n = idx - B2_OFF;
            if (n < OUT_F) val = b2[n];
        }
        ws[idx] = val;
    }
}

// ---------------------------------------------------------------------------
// Main: one wave per 16-character tile, two WMMA GEMMs back to back.
// ---------------------------------------------------------------------------
__global__ void __launch_bounds__(WAVES * 32)
enc_main_kernel(const float* __restrict__ x, const float* __restrict__ ws,
                float* __restrict__ out, int Bn) {
    extern __shared__ float smem[];
    const int tid  = threadIdx.x;
    const int lane = tid & 31;
    const int wave = tid >> 5;

    // cooperative weight load (coalesced b32 stream, ~90 KB)
    for (int i = tid; i < WS_FLOATS; i += blockDim.x) smem[i] = ws[i];
    __syncthreads();

    const float* w1s = smem;
    const float* w2s = smem + W1S_FLOATS;
    const float* c1  = smem + C1_OFF;
    const float* b2s = smem + B2_OFF;
    float* stage     = smem + WS_FLOATS + wave * STAGE_FLOATS;

    const int charBase = blockIdx.x * CHARS_PER_BLOCK + wave * 16;
    if (charBase >= Bn) return;            // wave-uniform: EXEC stays all-1s

    const int M    = lane & 15;            // char row within tile
    const int koff = (lane >> 4) << 1;     // 0 (lanes 0-15) or 2 (lanes 16-31)
    const int mAdd = (lane >> 4) << 3;     // D-layout row offset: 0 or 8
    const int row  = charBase + M;
    const int rowc = row < Bn ? row : (Bn - 1);
    const bool rowValid = row < Bn;

    const float* xrow  = x   + (size_t)rowc * IN_F;
    float*       oxrow = out + (size_t)rowc * IN_F;
    float* out1g = out + (size_t)Bn * IN_F;
    float* out2g = out + (size_t)Bn * (IN_F + P_F);

    // ---- load A fragments for GEMM1 directly in WMMA layout (covers tile once)
    v2f a[KSTEPS];
#pragma unroll
    for (int k = 0; k < KSTEPS; ++k)
        a[k] = *(const v2f*)(xrow + 4 * k + koff);

    // ---- pass-through copy of x into d_out[0 : Bn*84] from the same registers
    if (rowValid) {
#pragma unroll
        for (int k = 0; k < KSTEPS; ++k)
            *(v2f*)(oxrow + 4 * k + koff) = a[k];
    }

    // ======================= GEMM1: [16x84] @ [84x96] ======================
    for (int t = 0; t < NT1; ++t) {
        v8f acc = {0.f, 0.f, 0.f, 0.f, 0.f, 0.f, 0.f, 0.f};
#pragma unroll
        for (int k = 0; k < KSTEPS; ++k) {
            v2f b = *(const v2f*)(w1s + (t * KSTEPS + k) * 64 + lane * 2);
            acc = __builtin_amdgcn_wmma_f32_16x16x4_f32(
                false, a[k], false, b, (short)0, acc, false, false);
        }
        const int n = 16 * t + (lane & 15);
        const float bias = c1[n];
#pragma unroll
        for (int j = 0; j < 8; ++j) {
            float v = leaky(acc[j] + bias);
            stage[(j + mAdd) * STAGE_STRIDE + n] = v;      // for GEMM2 transpose
            int ch = charBase + j + mAdd;
            if (n < P_F && ch < Bn)
                out1g[(size_t)ch * P_F + n] = v;
        }
    }

    // ---- reload out1 tile as A fragments (cross-lane transpose via LDS)
#pragma unroll
    for (int k = 0; k < KSTEPS; ++k)
        a[k] = *(const v2f*)(stage + M * STAGE_STRIDE + 4 * k + koff);

    // ====================== GEMM2: [16x84] @ [84x176] ======================
    for (int t = 0; t < NT2; ++t) {
        v8f acc = {0.f, 0.f, 0.f, 0.f, 0.f, 0.f, 0.f, 0.f};
#pragma unroll
        for (int k = 0; k < KSTEPS; ++k) {
            v2f b = *(const v2f*)(w2s + (t * KSTEPS + k) * 64 + lane * 2);
            acc = __builtin_amdgcn_wmma_f32_16x16x4_f32(
                false, a[k], false, b, (short)0, acc, false, false);
        }
        const int n = 16 * t + (lane & 15);
        const float bias = b2s[n];
#pragma unroll
        for (int j = 0; j < 8; ++j) {
            float v = leaky(acc[j] + bias);
            int ch = charBase + j + mAdd;
            if (n < OUT_F && ch < Bn)
                out2g[(size_t)ch * OUT_F + n] = v;
        }
    }
}

// ---------------------------------------------------------------------------
extern "C" void kernel_launch(void* const* d_in, const int* in_sizes, int n_in,
                              void* d_out, int out_size, void* d_ws, size_t ws_size,
                              hipStream_t stream) {
    (void)n_in; (void)out_size; (void)ws_size;
    const float* x      = (const float*)d_in[0];
    const float* W1     = (const float*)d_in[1];
    const float* b1     = (const float*)d_in[2];
    const float* W2     = (const float*)d_in[3];
    const float* b2     = (const float*)d_in[4];
    const float* mask1  = (const float*)d_in[5];
    const float* mask2  = (const float*)d_in[6];
    const float* pool_w = (const float*)d_in[7];
    float* out = (float*)d_out;
    float* ws  = (float*)d_ws;

    const int Bn = in_sizes[0] / IN_F;

    enc_prep_kernel<<<32, 256, 0, stream>>>(W1, b1, W2, b2, mask1, mask2, pool_w, ws);

    const int blocks = (Bn + CHARS_PER_BLOCK - 1) / CHARS_PER_BLOCK;
    const size_t smemBytes = (size_t)SMEM_FLOATS * sizeof(float);
    enc_main_kernel<<<blocks, WAVES * 32, smemBytes, stream>>>(x, ws, out, Bn);
}